// TestModule_18064632447372
// MI455X (gfx1250) — compile-verified
//
#include <hip/hip_runtime.h>
#include <math.h>

typedef __attribute__((ext_vector_type(2))) float v2f;
typedef __attribute__((ext_vector_type(8))) float v8f;

#define N_NODES 100000
#define N_EDGES 3200000
#define D_FEAT  64
#define H_SIZE  32
#define NC      10
#define NCP     16   // classes padded to 16 for WMMA / coalescing

// ---------------------------------------------------------------------------
// zero fill
// ---------------------------------------------------------------------------
__global__ void zero_f32(float* __restrict__ p, int n) {
    int i = blockIdx.x * blockDim.x + threadIdx.x;
    if (i < n) p[i] = 0.0f;
}

// ---------------------------------------------------------------------------
// Layer 1 transform: z1 = x @ W1_rel^T   (N x 32)
//                    r1 = x @ W1_root^T + b1  (N x 32)
// One wave32 per 16-node tile. 64 x v_wmma_f32_16x16x4_f32 per wave.
// ---------------------------------------------------------------------------
__launch_bounds__(256)
__global__ void l1_transform(const float* __restrict__ x,
                             const float* __restrict__ W_rel,   // [32,64]
                             const float* __restrict__ b1,      // [32]
                             const float* __restrict__ W_root,  // [32,64]
                             float* __restrict__ z1,            // [N,32]
                             float* __restrict__ r1)            // [N,32]
{
    const int lane = threadIdx.x & 31;
    const int wave = (blockIdx.x * (blockDim.x >> 5)) + (threadIdx.x >> 5);
    const int base = wave * 16;
    if (base >= N_NODES) return;

    const int mrow = lane & 15;            // A row within tile
    const int koff = (lane >> 4) * 2;      // 0 or 2 (K pair select)
    const int ncol = lane & 15;            // B/D column within N-tile

    const int row  = min(base + mrow, N_NODES - 1);  // clamp: keep EXEC uniform
    const float rowvalid = (base + mrow < N_NODES) ? 1.0f : 0.0f;

    v8f acc_rel0 = {}, acc_rel1 = {}, acc_rt0 = {}, acc_rt1 = {};

    #pragma unroll
    for (int kk = 0; kk < 16; ++kk) {
        const int k = kk * 4 + koff;

        v2f a;
        a.x = x[row * D_FEAT + k]     * rowvalid;
        a.y = x[row * D_FEAT + k + 1] * rowvalid;

        v2f br0, br1, bt0, bt1;                 // B[k][n] = W[n][k]
        br0.x = W_rel [(ncol     ) * D_FEAT + k];
        br0.y = W_rel [(ncol     ) * D_FEAT + k + 1];
        br1.x = W_rel [(ncol + 16) * D_FEAT + k];
        br1.y = W_rel [(ncol + 16) * D_FEAT + k + 1];
        bt0.x = W_root[(ncol     ) * D_FEAT + k];
        bt0.y = W_root[(ncol     ) * D_FEAT + k + 1];
        bt1.x = W_root[(ncol + 16) * D_FEAT + k];
        bt1.y = W_root[(ncol + 16) * D_FEAT + k + 1];

        acc_rel0 = __builtin_amdgcn_wmma_f32_16x16x4_f32(false, a, false, br0,
                                                         (short)0, acc_rel0, false, false);
        acc_rel1 = __builtin_amdgcn_wmma_f32_16x16x4_f32(false, a, false, br1,
                                                         (short)0, acc_rel1, false, false);
        acc_rt0  = __builtin_amdgcn_wmma_f32_16x16x4_f32(false, a, false, bt0,
                                                         (short)0, acc_rt0, false, false);
        acc_rt1  = __builtin_amdgcn_wmma_f32_16x16x4_f32(false, a, false, bt1,
                                                         (short)0, acc_rt1, false, false);
    }

    const float bias0 = b1[ncol];
    const float bias1 = b1[ncol + 16];
    const int   mhalf = (lane >> 4) * 8;   // D layout: upper lanes hold M+8

    #pragma unroll
    for (int g = 0; g < 8; ++g) {
        const int node = base + g + mhalf;
        if (node < N_NODES) {
            z1[node * H_SIZE + ncol]      = acc_rel0[g];
            z1[node * H_SIZE + 16 + ncol] = acc_rel1[g];
            r1[node * H_SIZE + ncol]      = acc_rt0[g] + bias0;
            r1[node * H_SIZE + 16 + ncol] = acc_rt1[g] + bias1;
        }
    }
}

// ---------------------------------------------------------------------------
// Edge scatter, layer 1: agg1[dst] += z1[src].  One wave per edge (32 chans).
// Wave-uniform index loads, coalesced 128B gather, contiguous f32 atomics (L2).
// ---------------------------------------------------------------------------
__global__ void scatter1(const int* __restrict__ ei,   // [2, E]
                         const float* __restrict__ z1,
                         float* __restrict__ agg1)
{
    const unsigned t = blockIdx.x * blockDim.x + threadIdx.x;
    const int e = (int)(t >> 5);
    const int c = (int)(t & 31u);
    if (e >= N_EDGES) return;
    const int src = ei[e];
    const int dst = ei[N_EDGES + e];
    atomicAdd(&agg1[dst * H_SIZE + c], z1[src * H_SIZE + c]);
}

// ---------------------------------------------------------------------------
// Layer 2 transform: h = relu(agg1 + r1) (kept in registers, never stored)
//                    z2 = h @ W2_rel^T           (N x 16, cols>=10 == 0)
//                    r2 = h @ W2_root^T + b2     (N x 16)
// One wave32 per 16-node tile. 16 x v_wmma_f32_16x16x4_f32 per wave.
// ---------------------------------------------------------------------------
__launch_bounds__(256)
__global__ void l2_transform(const float* __restrict__ agg1,
                             const float* __restrict__ r1,
                             const float* __restrict__ W2_rel,  // [10,32]
                             const float* __restrict__ b2,      // [10]
                             const float* __restrict__ W2_root, // [10,32]
                             float* __restrict__ z2,            // [N,16]
                             float* __restrict__ r2)            // [N,16]
{
    const int lane = threadIdx.x & 31;
    const int wave = (blockIdx.x * (blockDim.x >> 5)) + (threadIdx.x >> 5);
    const int base = wave * 16;
    if (base >= N_NODES) return;

    const int mrow = lane & 15;
    const int koff = (lane >> 4) * 2;
    const int ncol = lane & 15;

    const int   row      = min(base + mrow, N_NODES - 1);
    const float rowvalid = (base + mrow < N_NODES) ? 1.0f : 0.0f;
    const int   wc       = min(ncol, NC - 1);            // clamp weight row
    const float cmask    = (ncol < NC) ? 1.0f : 0.0f;    // zero padded cols

    v8f acc_rel = {}, acc_rt = {};

    #pragma unroll
    for (int kk = 0; kk < 8; ++kk) {
        const int k = kk * 4 + koff;

        v2f a;   // h fragment = relu(agg1 + r1)
        a.x = fmaxf(agg1[row * H_SIZE + k]     + r1[row * H_SIZE + k],     0.0f) * rowvalid;
        a.y = fmaxf(agg1[row * H_SIZE + k + 1] + r1[row * H_SIZE + k + 1], 0.0f) * rowvalid;

        v2f brel, brt;
        brel.x = W2_rel [wc * H_SIZE + k]     * cmask;
        brel.y = W2_rel [wc * H_SIZE + k + 1] * cmask;
        brt.x  = W2_root[wc * H_SIZE + k]     * cmask;
        brt.y  = W2_root[wc * H_SIZE + k + 1] * cmask;

        acc_rel = __builtin_amdgcn_wmma_f32_16x16x4_f32(false, a, false, brel,
                                                        (short)0, acc_rel, false, false);
        acc_rt  = __builtin_amdgcn_wmma_f32_16x16x4_f32(false, a, false, brt,
                                                        (short)0, acc_rt, false, false);
    }

    const float bias  = (ncol < NC) ? b2[wc] : 0.0f;
    const int   mhalf = (lane >> 4) * 8;

    #pragma unroll
    for (int g = 0; g < 8; ++g) {
        const int node = base + g + mhalf;
        if (node < N_NODES) {
            z2[node * NCP + ncol] = acc_rel[g];
            r2[node * NCP + ncol] = acc_rt[g] + bias;
        }
    }
}

// ---------------------------------------------------------------------------
// Edge scatter, layer 2: agg2[dst] += z2[src] over the 10 real classes.
// Half-wave per edge; padded stride 16 keeps gathers coalesced.
// ---------------------------------------------------------------------------
__global__ void scatter2(const int* __restrict__ ei,
                         const float* __restrict__ z2,
                         float* __restrict__ agg2)
{
    const unsigned t = blockIdx.x * blockDim.x + threadIdx.x;
    const int e = (int)(t >> 4);
    const int c = (int)(t & 15u);
    if (e >= N_EDGES) return;
    const int src = ei[e];
    const int dst = ei[N_EDGES + e];
    if (c < NC) {
        atomicAdd(&agg2[dst * NCP + c], z2[src * NCP + c]);
    }
}

// ---------------------------------------------------------------------------
// Loss: logits = agg2 + r2 ; loss = mean(logsumexp(logits) - logits[y])
// Block-reduce in LDS, one atomicAdd per block into d_out (pre-zeroed).
// ---------------------------------------------------------------------------
__launch_bounds__(256)
__global__ void loss_kernel(const float* __restrict__ agg2,
                            const float* __restrict__ r2,
                            const int* __restrict__ y,
                            float* __restrict__ out)
{
    const int n = blockIdx.x * blockDim.x + threadIdx.x;
    float contrib = 0.0f;
    if (n < N_NODES) {
        float l[NC];
        float m = -1e30f;
        #pragma unroll
        for (int c = 0; c < NC; ++c) {
            l[c] = agg2[n * NCP + c] + r2[n * NCP + c];
            m = fmaxf(m, l[c]);
        }
        float s = 0.0f;
        #pragma unroll
        for (int c = 0; c < NC; ++c) s += expf(l[c] - m);
        const float lse = m + logf(s);
        const int   t   = y[n];
        contrib = (lse - l[t]) * (1.0f / (float)N_NODES);
    }

    __shared__ float red[256];
    red[threadIdx.x] = contrib;
    __syncthreads();
    #pragma unroll
    for (int s = 128; s > 0; s >>= 1) {
        if ((int)threadIdx.x < s) red[threadIdx.x] += red[threadIdx.x + s];
        __syncthreads();
    }
    if (threadIdx.x == 0) atomicAdd(out, red[0]);
}

// ---------------------------------------------------------------------------
// launch
// ---------------------------------------------------------------------------
extern "C" void kernel_launch(void* const* d_in, const int* in_sizes, int n_in,
                              void* d_out, int out_size, void* d_ws, size_t ws_size,
                              hipStream_t stream) {
    const float* x       = (const float*)d_in[0];
    const int*   ei      = (const int*)  d_in[1];   // [2, E] int32
    const int*   y       = (const int*)  d_in[2];
    const float* W1_rel  = (const float*)d_in[3];
    const float* b1      = (const float*)d_in[4];
    const float* W1_root = (const float*)d_in[5];
    const float* W2_rel  = (const float*)d_in[6];
    const float* b2      = (const float*)d_in[7];
    const float* W2_root = (const float*)d_in[8];
    float* out = (float*)d_out;

    float* ws   = (float*)d_ws;
    float* z1   = ws;                               // N*32
    float* r1   = z1   + (size_t)N_NODES * H_SIZE;  // N*32
    float* agg1 = r1   + (size_t)N_NODES * H_SIZE;  // N*32
    float* z2   = agg1 + (size_t)N_NODES * H_SIZE;  // N*16
    float* r2   = z2   + (size_t)N_NODES * NCP;     // N*16
    float* agg2 = r2   + (size_t)N_NODES * NCP;     // N*16
    // total: 57.6 MB of workspace

    const int nt = 256;

    // zero accumulators + output
    zero_f32<<<(N_NODES * H_SIZE + nt - 1) / nt, nt, 0, stream>>>(agg1, N_NODES * H_SIZE);
    zero_f32<<<(N_NODES * NCP    + nt - 1) / nt, nt, 0, stream>>>(agg2, N_NODES * NCP);
    zero_f32<<<1, 32, 0, stream>>>(out, 1);

    // layer 1 transform (one wave per 16 nodes; 8 waves per block)
    const int n_tiles  = (N_NODES + 15) / 16;            // 6250
    const int t_blocks = (n_tiles + 7) / 8;              // 782
    l1_transform<<<t_blocks, nt, 0, stream>>>(x, W1_rel, b1, W1_root, z1, r1);

    // layer 1 scatter: one wave per edge
    {
        const long long total = (long long)N_EDGES * 32;
        scatter1<<<(unsigned)((total + nt - 1) / nt), nt, 0, stream>>>(ei, z1, agg1);
    }

    // layer 2 transform (fuses relu(agg1+r1) -> h in registers)
    l2_transform<<<t_blocks, nt, 0, stream>>>(agg1, r1, W2_rel, b2, W2_root, z2, r2);

    // layer 2 scatter: half-wave per edge
    {
        const long long total = (long long)N_EDGES * NCP;
        scatter2<<<(unsigned)((total + nt - 1) / nt), nt, 0, stream>>>(ei, z2, agg2);
    }

    // loss
    loss_kernel<<<(N_NODES + nt - 1) / nt, nt, 0, stream>>>(agg2, r2, y, out);
}